// TorchAccelerator_63831803953170
// MI455X (gfx1250) — compile-verified
//
#include <hip/hip_runtime.h>
#include <hip/hip_bf16.h>

// ---------------------------------------------------------------------------
// Chain product of 500k 6x6 f32 matrices, then M @ R @ M^T.
// Each 6x6 left-multiply = two chained V_WMMA_F32_16X16X4_F32 (K = 4+2, zero
// padded). Associative 3-level reduction: 4096 segments (2 per wave for ILP)
// -> 64 -> 1. Bandwidth floor: 72 MB / 23.3 TB/s ~= 3.1 us.
// A-layout gathers use 8-byte-aligned global_load_b64 (adjacent K pairs).
// ---------------------------------------------------------------------------

typedef float v2f __attribute__((ext_vector_type(2)));
typedef float v8f __attribute__((ext_vector_type(8)));

struct Amat { v2f a1; v2f a2; };   // A operands for the two K-slices
struct Bmat { v2f b1; v2f b2; };   // B operands for the two K-slices

// Per-lane gather offsets (elements, 8B-aligned pair bases) + value masks.
struct LaneOffs { int o01, o23; float m01, m23; };

__device__ __forceinline__ int laneId() { return (int)(threadIdx.x & 31); }

__device__ __forceinline__ LaneOffs makeOffs() {
    int lane = laneId();
    int r    = lane & 15;
    bool hi  = lane >= 16;
    bool rv  = r < 6;          // row valid (both lane halves)
    bool lv  = !hi && rv;      // low-half rows carry K=4,5
    LaneOffs L;
    L.o01 = rv ? r * 6 + (hi ? 2 : 0) : 0;   // pair (K0,K1)/(K2,K3): even index
    L.o23 = lv ? r * 6 + 4 : 0;              // pair (K4,K5): even index
    L.m01 = rv ? 1.0f : 0.0f;
    L.m23 = lv ? 1.0f : 0.0f;
    return L;
}

// Gather a row-major 6x6 into WMMA A-layout (16x4 f32) with two b64 loads:
//   .x VGPR: K=0 lanes 0-15 / K=2 lanes 16-31 ; .y VGPR: K=1 / K=3.
// Zero-padding via float multiply (0 * finite == 0), never via branches.
__device__ __forceinline__ Amat loadA(const float* __restrict__ p, const LaneOffs& L) {
    v2f x01 = *(const v2f*)(p + L.o01);
    v2f x23 = *(const v2f*)(p + L.o23);
    Amat A;
    A.a1.x = x01.x * L.m01;
    A.a1.y = x01.y * L.m01;
    A.a2.x = x23.x * L.m23;
    A.a2.y = x23.y * L.m23;
    return A;
}

// Gather a row-major 6x6 (optionally transposed) into WMMA B-layout
// (4x16 f32): .x: K=0 lanes 0-15 / K=2 lanes 16-31 ; .y: K=1 / K=3.
// Cold path (used twice in phase 3), plain b32 gathers.
__device__ __forceinline__ Bmat loadB(const float* __restrict__ m, bool transpose) {
    int lane = laneId();
    int n    = lane & 15;
    bool hi  = lane >= 16;
    bool nv  = n < 6;
    bool lv  = !hi && nv;
    int k0 = hi ? 2 : 0;
    int k1 = hi ? 3 : 1;
    int i0 = transpose ? n * 6 + k0 : k0 * 6 + n;
    int i1 = transpose ? n * 6 + k1 : k1 * 6 + n;
    int i2 = transpose ? n * 6 + 4  : 4 * 6 + n;
    int i3 = transpose ? n * 6 + 5  : 5 * 6 + n;
    float w01 = nv ? 1.0f : 0.0f;
    float w23 = lv ? 1.0f : 0.0f;
    Bmat B;
    B.b1.x = m[nv ? i0 : 0] * w01;
    B.b1.y = m[nv ? i1 : 0] * w01;
    B.b2.x = m[lv ? i2 : 0] * w23;
    B.b2.y = m[lv ? i3 : 0] * w23;
    return B;
}

// D = A(16x6 padded) x B(6x16 padded) via two chained f32 WMMAs.
__device__ __forceinline__ v8f mm6(const Amat& A, const Bmat& B) {
    v8f zero = {};
    v8f t = __builtin_amdgcn_wmma_f32_16x16x4_f32(
        false, A.a1, false, B.b1, (short)0, zero, false, false);
    return __builtin_amdgcn_wmma_f32_16x16x4_f32(
        false, A.a2, false, B.b2, (short)0, t, false, false);
}

// C/D layout (row r in VGPR r, lanes 0-15) -> B layout for the next
// left-multiply. Rows >=6 / cols >=6 of the tile are identically zero.
__device__ __forceinline__ Bmat dToB(v8f c) {
    int lane = laneId();
    bool hi  = lane >= 16;
    int src  = lane & 15;
    float r2 = __shfl(c[2], src, 32);
    float r3 = __shfl(c[3], src, 32);
    Bmat B;
    B.b1.x = hi ? r2 : c[0];
    B.b1.y = hi ? r3 : c[1];
    B.b2.x = hi ? 0.0f : c[4];
    B.b2.y = hi ? 0.0f : c[5];
    return B;
}

// 6x6 identity in C/D layout.
__device__ __forceinline__ v8f identD() {
    int lane = laneId();
    int n    = lane & 15;
    bool hi  = lane >= 16;
    v8f c = {};
#pragma unroll
    for (int r = 0; r < 8; ++r) {
        int row = hi ? r + 8 : r;
        c[r] = (row == n && n < 6) ? 1.0f : 0.0f;
    }
    return c;
}

// Store the 6x6 block of a C/D-layout tile to row-major memory.
__device__ __forceinline__ void storeD(v8f c, float* __restrict__ out) {
    int lane = laneId();
    if (lane < 6) {
        out[0 * 6 + lane] = c[0];
        out[1 * 6 + lane] = c[1];
        out[2 * 6 + lane] = c[2];
        out[3 * 6 + lane] = c[3];
        out[4 * 6 + lane] = c[4];
        out[5 * 6 + lane] = c[5];
    }
}

// ---------------------------------------------------------------------------
// Phase 1: each wave folds TWO independent contiguous segments (2x ILP on the
// wmma->bpermute->wmma critical path), writing two ordered partials.
// ---------------------------------------------------------------------------
__global__ void accel_chain_phase1(const float* __restrict__ E,
                                   float* __restrict__ partials,
                                   int N, int segLen, int nChain) {
    int wave = (int)((blockIdx.x * blockDim.x + threadIdx.x) >> 5);
    int ch0 = wave * 2;
    int ch1 = ch0 + 1;
    if (ch0 >= nChain) return;

    const LaneOffs L = makeOffs();

    int s0 = ch0 * segLen; if (s0 > N) s0 = N;
    int e0 = s0 + segLen;  if (e0 > N) e0 = N;
    int s1 = ch1 * segLen; if (s1 > N) s1 = N;
    int e1 = s1 + segLen;  if (e1 > N) e1 = N;

    int n0 = e0 - s0;
    int n1 = e1 - s1;
    int nc = n0 < n1 ? n0 : n1;

    const float* __restrict__ p0 = E + (size_t)s0 * 36;
    const float* __restrict__ p1 = E + (size_t)s1 * 36;

    v8f ca = identD();
    v8f cb = identD();

    // Joint loop: two independent dependency chains in flight.
    for (int i = 0; i < nc; ++i) {
        Bmat Ba = dToB(ca);
        Bmat Bb = dToB(cb);
        Amat Aa = loadA(p0, L);
        Amat Ab = loadA(p1, L);
        ca = mm6(Aa, Ba);
        cb = mm6(Ab, Bb);
        p0 += 36;
        p1 += 36;
    }
    // Remainders (at most one of these runs, only for the ragged tail).
    for (int i = nc; i < n0; ++i) {
        Bmat B = dToB(ca);
        Amat A = loadA(p0, L);
        ca = mm6(A, B);
        p0 += 36;
    }
    for (int i = nc; i < n1; ++i) {
        Bmat B = dToB(cb);
        Amat A = loadA(p1, L);
        cb = mm6(A, B);
        p1 += 36;
    }

    storeD(ca, partials + (size_t)ch0 * 36);
    storeD(cb, partials + (size_t)ch1 * 36);
}

// ---------------------------------------------------------------------------
// Phase 2: each wave folds one group of partials in order.
// ---------------------------------------------------------------------------
__global__ void accel_chain_phase2(const float* __restrict__ P,
                                   float* __restrict__ Q,
                                   int nIn, int group, int nOut) {
    int wave = (int)((blockIdx.x * blockDim.x + threadIdx.x) >> 5);
    if (wave >= nOut) return;

    const LaneOffs L = makeOffs();
    int start = wave * group;
    int end   = start + group;
    if (end > nIn) end = nIn;

    const float* __restrict__ p = P + (size_t)start * 36;
    v8f c = identD();
    for (int s = start; s < end; ++s) {
        Bmat B = dToB(c);
        Amat A = loadA(p, L);
        c = mm6(A, B);
        p += 36;
    }
    storeD(c, Q + (size_t)wave * 36);
}

// ---------------------------------------------------------------------------
// Phase 3 (single wave): final fold, then out = M @ R @ M^T.
// ---------------------------------------------------------------------------
__global__ void accel_chain_phase3(const float* __restrict__ Q, int nQ,
                                   const float* __restrict__ R,
                                   float* __restrict__ out,
                                   float* __restrict__ scratch) {
    const LaneOffs L = makeOffs();

    v8f c = identD();
    const float* __restrict__ p = Q;
    for (int s = 0; s < nQ; ++s) {
        Bmat B = dToB(c);
        Amat A = loadA(p, L);
        c = mm6(A, B);
        p += 36;
    }

    // M (D-layout) -> row-major scratch so it can be re-striped as A / B^T.
    storeD(c, scratch);
    asm volatile("s_wait_storecnt 0x0" ::: "memory");

    // T1 = M @ R
    Amat Am = loadA(scratch, L);
    Bmat Br = loadB(R, /*transpose=*/false);
    v8f t1 = mm6(Am, Br);

    storeD(t1, scratch + 64);
    asm volatile("s_wait_storecnt 0x0" ::: "memory");

    // out = T1 @ M^T
    Amat At = loadA(scratch + 64, L);
    Bmat Bm = loadB(scratch, /*transpose=*/true);
    v8f o = mm6(At, Bm);

    storeD(o, out);
}

// ---------------------------------------------------------------------------
// Host-side launcher
// ---------------------------------------------------------------------------
extern "C" void kernel_launch(void* const* d_in, const int* in_sizes, int n_in,
                              void* d_out, int out_size, void* d_ws, size_t ws_size,
                              hipStream_t stream) {
    const float* E = (const float*)d_in[0];   // [N, 6, 6] f32
    const float* R = (const float*)d_in[1];   // [6, 6]    f32
    float* out     = (float*)d_out;           // [6, 6]    f32

    const int N = in_sizes[0] / 36;

    const int nChain = 4096;                         // phase-1 segments
    const int group  = 64;                           // phase-2 fan-in
    const int nQ     = (nChain + group - 1) / group; // = 64

    int segLen = (N + nChain - 1) / nChain;
    if (segLen < 1) segLen = 1;

    float* ws       = (float*)d_ws;
    float* partials = ws;                              // nChain * 36 floats
    float* q        = ws + (size_t)nChain * 36;        // nQ * 36 floats
    float* scratch  = ws + (size_t)(nChain + nQ) * 36; // 128 floats

    // Phase 1: 256 blocks x 256 threads = 2048 waves, 4096 chains.
    accel_chain_phase1<<<dim3(256), dim3(256), 0, stream>>>(
        E, partials, N, segLen, nChain);

    // Phase 2: 8 blocks x 256 threads = 64 waves, each folds 64 partials.
    accel_chain_phase2<<<dim3(8), dim3(256), 0, stream>>>(
        partials, q, nChain, group, nQ);

    // Phase 3: one wave folds the 64 group products and applies R.
    accel_chain_phase3<<<dim3(1), dim3(32), 0, stream>>>(
        q, nQ, R, out, scratch);
}